// SentimentClassificationBlock_18528488915008
// MI455X (gfx1250) — compile-verified
//
#include <hip/hip_runtime.h>
#include <hip/hip_bf16.h>

// ---------------------------------------------------------------------------
// Problem constants (from reference)
// ---------------------------------------------------------------------------
#define Bb 8
#define Ss_ 512
#define Hh 768
#define Aa 8
#define NHh 12
#define DHh 64
#define Pp 256
#define MAXT 32
#define NCLS 3

typedef __attribute__((ext_vector_type(2))) float v2f;
typedef __attribute__((ext_vector_type(8))) float v8f;

__device__ __forceinline__ v2f ld2(const float* p) {
    return *reinterpret_cast<const v2f*>(p);
}

// V_WMMA_F32_16X16X4_F32 : D(16x16 f32) = A(16x4 f32) * B(4x16 f32) + C
__device__ __forceinline__ v8f wmma_f32(v2f a, v2f b, v8f c) {
    return __builtin_amdgcn_wmma_f32_16x16x4_f32(
        false, a, false, b, (short)0, c, false, false);
}

// ---------------------------------------------------------------------------
// Generic tiled GEMM: C[M,N] = act((A[M,K] @ W[K,N] + bias) * scale)
// 256 threads = 8 waves; block tile 64x128, wave tile 32x32 (2x2 WMMA accs).
// Double-buffered LDS (one barrier per K-tile); W tile stored TRANSPOSED so
// both A and B fragments are single aligned 8-byte ds loads.
// Requires M % 64 == 0, N % 128 == 0, K % 32 == 0 (true for all uses here).
// ---------------------------------------------------------------------------
#define BM 64
#define BN 128
#define BK 32
#define KSTR 36   // BK + 4 pad (even -> float2-aligned, avoids bank conflicts)

__global__ __launch_bounds__(256) void gemm_bias_act(
    const float* __restrict__ A, const float* __restrict__ W,
    const float* __restrict__ bias, float* __restrict__ C,
    int M, int N, int K, float scale, int relu)
{
    __shared__ __align__(16) float As[2][BM][KSTR];
    __shared__ __align__(16) float WtT[2][BN][KSTR];   // transposed: [n][k]

    const int tid  = threadIdx.x;
    const int wave = tid >> 5;
    const int lane = tid & 31;
    const int hf   = lane >> 4;     // half-wave (0/1)
    const int lr   = lane & 15;
    const int m0   = blockIdx.x * BM;
    const int n0   = blockIdx.y * BN;
    const int wm   = (wave & 1) * 32;
    const int wn   = (wave >> 1) * 32;

    if (m0 >= M) return;

    const int ar  = tid >> 3;        // A-tile row 0..31 (and +32)
    const int ac4 = (tid & 7) * 4;   // A-tile col (float4)

    auto load_tile = [&](int k0, int buf) {
        const float* g = A + (size_t)(m0 + ar) * K + k0 + ac4;
        *(float4*)&As[buf][ar][ac4]      = *(const float4*)g;
        *(float4*)&As[buf][ar + 32][ac4] = *(const float4*)(g + (size_t)32 * K);
        for (int i = tid; i < (BK * BN) / 4; i += 256) {
            int r  = i >> 5;          // k row 0..31
            int c4 = (i & 31) * 4;    // n col
            float4 w = *(const float4*)&W[(size_t)(k0 + r) * N + n0 + c4];
            WtT[buf][c4 + 0][r] = w.x;
            WtT[buf][c4 + 1][r] = w.y;
            WtT[buf][c4 + 2][r] = w.z;
            WtT[buf][c4 + 3][r] = w.w;
        }
    };

    v8f acc[2][2] = {};
    load_tile(0, 0);
    const int nk = K / BK;
    for (int it = 0; it < nk; ++it) {
        const int cur = it & 1;
        __syncthreads();
        if (it + 1 < nk) load_tile((it + 1) * BK, cur ^ 1);
        #pragma unroll
        for (int kk = 0; kk < BK; kk += 4) {
            v2f a0 = ld2(&As[cur][wm + lr][kk + 2 * hf]);
            v2f a1 = ld2(&As[cur][wm + 16 + lr][kk + 2 * hf]);
            v2f b0 = ld2(&WtT[cur][wn + lr][kk + 2 * hf]);
            v2f b1 = ld2(&WtT[cur][wn + 16 + lr][kk + 2 * hf]);
            acc[0][0] = wmma_f32(a0, b0, acc[0][0]);
            acc[0][1] = wmma_f32(a0, b1, acc[0][1]);
            acc[1][0] = wmma_f32(a1, b0, acc[1][0]);
            acc[1][1] = wmma_f32(a1, b1, acc[1][1]);
        }
    }

    #pragma unroll
    for (int ti = 0; ti < 2; ++ti) {
        #pragma unroll
        for (int tj = 0; tj < 2; ++tj) {
            int gn = n0 + wn + tj * 16 + lr;
            float bv = bias ? bias[gn] : 0.0f;
            #pragma unroll
            for (int r = 0; r < 8; ++r) {
                int gm  = m0 + wm + ti * 16 + r + 8 * hf;
                float v = acc[ti][tj][r] + bv;
                if (relu) v = fmaxf(v, 0.0f);
                C[(size_t)gm * N + gn] = v * scale;
            }
        }
    }
}

// ---------------------------------------------------------------------------
// merge[p, t, h] = all_tok[bidx[p], clip(t0+t)][h] + mean_j(asp_tok[bidx[p], a0+j][h])
// ---------------------------------------------------------------------------
__global__ __launch_bounds__(256) void build_merge(
    const float* __restrict__ all_tok, const float* __restrict__ asp_tok,
    const int* __restrict__ pos, const int* __restrict__ bidx_arr,
    float* __restrict__ merge)
{
    const int p   = blockIdx.x;
    const int tid = threadIdx.x;
    const int b   = bidx_arr[p];
    const int t0  = pos[p * 4 + 0];
    const int a0  = pos[p * 4 + 2];
    const int a1  = pos[p * 4 + 3];
    const int alen = a1 - a0;
    const float inv = 1.0f / (float)alen;

    __shared__ float pool[Hh];
    for (int h = tid; h < Hh; h += 256) {
        float s = 0.0f;
        for (int j = 0; j < alen; ++j)
            s += asp_tok[((size_t)b * Aa + a0 + j) * Hh + h];
        pool[h] = s * inv;
    }
    __syncthreads();

    float* mout = merge + (size_t)p * MAXT * Hh;
    for (int i = tid; i < MAXT * Hh; i += 256) {
        int t = i / Hh;
        int h = i - t * Hh;
        int row = t0 + t;
        if (row > Ss_ - 1) row = Ss_ - 1;
        mout[i] = all_tok[((size_t)b * Ss_ + row) * Hh + h] + pool[h];
    }
}

// ---------------------------------------------------------------------------
// Flash attention per (p, head). 256 threads = 8 waves.
// Q: 32x64 resident (already pre-scaled); K/V streamed in 64-row chunks with
// online softmax. K chunk stored row-major [s][d]; V chunk stored TRANSPOSED
// [d][s] into the same buffer, so both QK^T and P*V fragments are single ld2s.
// LDS ~37 KB.
// ---------------------------------------------------------------------------
#define SCHUNK 64
#define QS_STR 68   // 64 + 4 pad
#define KV_STR 68
#define SC_STR 68

__global__ __launch_bounds__(256) void attention_kernel(
    const float* __restrict__ qlayer, const float* __restrict__ kall,
    const float* __restrict__ alltok, const float* __restrict__ amask,
    const int* __restrict__ bidx_arr, float* __restrict__ attn_out)
{
    const int p = blockIdx.x;
    const int n = blockIdx.y;
    const int b = bidx_arr[p];

    __shared__ __align__(16) float Qs[MAXT][QS_STR];
    __shared__ __align__(16) float KVs[SCHUNK][KV_STR];  // K: [s][d], V: [d][s]
    __shared__ __align__(16) float Sc[MAXT][SC_STR];
    __shared__ float Ms[Ss_];
    __shared__ float rowM[MAXT], rowL[MAXT], rowA[MAXT];

    const int tid  = threadIdx.x;
    const int wave = tid >> 5;
    const int lane = tid & 31;
    const int hf   = lane >> 4;
    const int lr   = lane & 15;
    const int ti   = wave >> 2;   // row-tile (0..1)
    const int cj   = wave & 3;    // col-tile (0..3) for S-chunk and O

    for (int i = tid * 4; i < MAXT * DHh; i += 1024) {
        int t = i >> 6, d = i & 63;
        *(float4*)&Qs[t][d] =
            *(const float4*)&qlayer[(((size_t)p * MAXT + t) * NHh + n) * DHh + d];
    }
    for (int i = tid; i < Ss_; i += 256)
        Ms[i] = amask[(size_t)b * Ss_ * Ss_ + i];
    if (tid < MAXT) { rowM[tid] = -3.0e38f; rowL[tid] = 0.0f; }

    v8f acc = {};  // this wave's 16x16 tile of O

    for (int sc = 0; sc < Ss_ / SCHUNK; ++sc) {
        // ---- load K chunk (row-major) ----
        __syncthreads();   // prev P*V finished with KVs
        for (int i = tid * 4; i < SCHUNK * DHh; i += 1024) {
            int s = i >> 6, d = i & 63;
            *(float4*)&KVs[s][d] = *(const float4*)
                &kall[((size_t)b * Ss_ + sc * SCHUNK + s) * Hh + n * DHh + d];
        }
        __syncthreads();

        // ---- S-chunk = Q (32x64) * K^T, per-wave 16x16 tile ----
        {
            v8f st = {};
            #pragma unroll
            for (int kk = 0; kk < DHh; kk += 4) {
                v2f a  = ld2(&Qs[ti * 16 + lr][kk + 2 * hf]);
                v2f bv = ld2(&KVs[cj * 16 + lr][kk + 2 * hf]);  // B[d][s] = K[s][d]
                st = wmma_f32(a, bv, st);
            }
            #pragma unroll
            for (int r = 0; r < 8; ++r)
                Sc[ti * 16 + r + 8 * hf][cj * 16 + lr] = st[r];
        }
        __syncthreads();

        // ---- online softmax update; each wave owns 4 rows ----
        #pragma unroll
        for (int rr = 0; rr < 4; ++rr) {
            int t = wave * 4 + rr;
            int s0 = lane, s1 = lane + 32;
            float x0 = Sc[t][s0] + (1.0f - Ms[sc * SCHUNK + s0]) * -10000.0f;
            float x1 = Sc[t][s1] + (1.0f - Ms[sc * SCHUNK + s1]) * -10000.0f;
            float mc = fmaxf(x0, x1);
            for (int o = 16; o > 0; o >>= 1) mc = fmaxf(mc, __shfl_xor(mc, o, 32));
            float m_old = rowM[t];
            float m_new = fmaxf(m_old, mc);
            float alpha = __expf(m_old - m_new);
            float p0 = __expf(x0 - m_new);
            float p1 = __expf(x1 - m_new);
            Sc[t][s0] = p0;
            Sc[t][s1] = p1;
            float ps = p0 + p1;
            for (int o = 16; o > 0; o >>= 1) ps += __shfl_xor(ps, o, 32);
            if (lane == 0) {
                rowL[t] = rowL[t] * alpha + ps;
                rowM[t] = m_new;
                rowA[t] = alpha;
            }
        }
        __syncthreads();

        // ---- rescale running O; load V chunk TRANSPOSED (disjoint LDS) ----
        #pragma unroll
        for (int r = 0; r < 8; ++r)
            acc[r] *= rowA[ti * 16 + r + 8 * hf];
        for (int i = tid * 4; i < SCHUNK * DHh; i += 1024) {
            int s = i >> 6, d = i & 63;
            float4 v = *(const float4*)
                &alltok[((size_t)b * Ss_ + sc * SCHUNK + s) * Hh + n * DHh + d];
            KVs[d + 0][s] = v.x;
            KVs[d + 1][s] = v.y;
            KVs[d + 2][s] = v.z;
            KVs[d + 3][s] = v.w;
        }
        __syncthreads();

        // ---- O += P (32x64) * V (64x64); B frag = single ld2 from V^T ----
        #pragma unroll
        for (int kk = 0; kk < SCHUNK; kk += 4) {
            v2f a  = ld2(&Sc[ti * 16 + lr][kk + 2 * hf]);
            v2f bv = ld2(&KVs[cj * 16 + lr][kk + 2 * hf]);  // V^T[d][s]
            acc = wmma_f32(a, bv, acc);
        }
    }

    // ---- normalize by l and write out: attn[p, t, n, d] ----
    #pragma unroll
    for (int r = 0; r < 8; ++r) {
        int t = ti * 16 + r + 8 * hf;
        float inv = 1.0f / rowL[t];
        attn_out[(((size_t)p * MAXT + t) * NHh + n) * DHh + cj * 16 + lr] =
            acc[r] * inv;
    }
}

// ---------------------------------------------------------------------------
// pooled[p] = mean over valid t of attn[p, t, :];  logits = pooled @ Wo + bo
// ---------------------------------------------------------------------------
__global__ __launch_bounds__(256) void pool_logits(
    const float* __restrict__ attn, const int* __restrict__ pos,
    const float* __restrict__ Wo, const float* __restrict__ bo,
    float* __restrict__ out)
{
    const int p    = blockIdx.x;
    const int tid  = threadIdx.x;
    const int tlen = pos[p * 4 + 1] - pos[p * 4 + 0];
    const float inv = 1.0f / (float)tlen;

    __shared__ float pooled[Hh];
    for (int h = tid; h < Hh; h += 256) {
        float s = 0.0f;
        for (int t = 0; t < tlen; ++t)
            s += attn[((size_t)p * MAXT + t) * Hh + h];
        pooled[h] = s * inv;
    }
    __syncthreads();

    const int wave = tid >> 5, lane = tid & 31;
    if (wave < NCLS) {
        float s = 0.0f;
        for (int h = lane; h < Hh; h += 32)
            s += pooled[h] * Wo[h * NCLS + wave];
        for (int o = 16; o > 0; o >>= 1) s += __shfl_xor(s, o, 32);
        if (lane == 0) out[p * NCLS + wave] = s + bo[wave];
    }
}

// ---------------------------------------------------------------------------
// Launch
// ---------------------------------------------------------------------------
extern "C" void kernel_launch(void* const* d_in, const int* in_sizes, int n_in,
                              void* d_out, int out_size, void* d_ws, size_t ws_size,
                              hipStream_t stream) {
    (void)in_sizes; (void)n_in; (void)out_size; (void)ws_size;

    const float* all_tokens    = (const float*)d_in[0];
    const float* aspect_tokens = (const float*)d_in[1];
    const float* amask         = (const float*)d_in[2];
    const int*   pos           = (const int*)d_in[3];
    const int*   bidx          = (const int*)d_in[4];
    const float* W1 = (const float*)d_in[5];
    const float* b1 = (const float*)d_in[6];
    const float* Wq = (const float*)d_in[7];
    const float* bq = (const float*)d_in[8];
    const float* Wk = (const float*)d_in[9];
    const float* bk = (const float*)d_in[10];
    const float* Wo = (const float*)d_in[11];
    const float* bo = (const float*)d_in[12];
    float* out = (float*)d_out;

    // Workspace layout (floats): ~101 MB total
    float* ws = (float*)d_ws;
    size_t o = 0;
    float* all_tok = ws + o; o += (size_t)Bb * Ss_ * Hh;
    float* asp_tok = ws + o; o += (size_t)Bb * Aa * Hh;
    float* k_all   = ws + o; o += (size_t)Bb * Ss_ * Hh;
    float* merge   = ws + o; o += (size_t)Pp * MAXT * Hh;
    float* qlayer  = ws + o; o += (size_t)Pp * MAXT * Hh;
    float* attn    = ws + o; o += (size_t)Pp * MAXT * Hh;

    dim3 blk(256);

    gemm_bias_act<<<dim3((Bb * Ss_) / BM, Hh / BN), blk, 0, stream>>>(
        all_tokens, W1, b1, all_tok, Bb * Ss_, Hh, Hh, 1.0f, 1);
    gemm_bias_act<<<dim3((Bb * Aa) / BM, Hh / BN), blk, 0, stream>>>(
        aspect_tokens, W1, b1, asp_tok, Bb * Aa, Hh, Hh, 1.0f, 1);
    gemm_bias_act<<<dim3((Bb * Ss_) / BM, Hh / BN), blk, 0, stream>>>(
        all_tok, Wk, bk, k_all, Bb * Ss_, Hh, Hh, 1.0f, 0);
    build_merge<<<dim3(Pp), blk, 0, stream>>>(all_tok, asp_tok, pos, bidx, merge);
    gemm_bias_act<<<dim3((Pp * MAXT) / BM, Hh / BN), blk, 0, stream>>>(
        merge, Wq, bq, qlayer, Pp * MAXT, Hh, Hh, 0.125f, 0);
    attention_kernel<<<dim3(Pp, NHh), blk, 0, stream>>>(
        qlayer, k_all, all_tok, amask, bidx, attn);
    pool_logits<<<dim3(Pp), blk, 0, stream>>>(attn, pos, Wo, bo, out);
}